// Model_60713657697040
// MI455X (gfx1250) — compile-verified
//
#include <hip/hip_runtime.h>
#include <cstdint>
#include <cstddef>

// ---------------------------------------------------------------------------
// LayerNorm backward (feature axis), B=4, S=4096, H=2048, fp32.
// Memory-bound: ~670MB traffic/call -> ~29us floor at 23.3 TB/s. Strategy:
//  - b128 non-temporal streaming loads/stores for dy/x1/x2/out
//  - async global->LDS staging of dsum (ASYNCcnt path, overlaps reduction)
//  - wave32 shuffle reduction + ping-pong LDS cross-wave combine (1 barrier/row)
//  - register-resident dgamma/dbeta partials, flushed via f32 global atomics
//  - no WMMA: op has no matmul structure; matrix pipes cannot reduce bytes
// ---------------------------------------------------------------------------

namespace {
constexpr int kB = 4;
constexpr int kS = 4096;
constexpr int kH = 2048;
constexpr int kRows = kB * kS;                   // 16384
constexpr int kBlock = 256;                      // 8 waves of 32
constexpr int kWaves = kBlock / 32;              // 8
constexpr int kCPT = kH / kBlock;                // 8 columns per thread
constexpr int kRowsPerBlock = 16;
constexpr int kGrid = kRows / kRowsPerBlock;     // 1024 blocks
constexpr float kInvD = 1.0f / (float)kH;
constexpr float kTwoInvD = 2.0f / (float)kH;
} // namespace

typedef float v4f __attribute__((ext_vector_type(4)));

__device__ __forceinline__ v4f nt_load4(const float* p) {
  return __builtin_nontemporal_load((const v4f*)p);
}
__device__ __forceinline__ void nt_store4(float* p, v4f v) {
  __builtin_nontemporal_store(v, (v4f*)p);
}

// Async 16B global -> LDS copy (CDNA5 ASYNCcnt path), non-temporal: dsum is
// read exactly once. The generic pointer to LDS carries the wave-relative LDS
// byte offset in its low 32 bits.
__device__ __forceinline__ void async_copy16(float* lds_dst, const float* g_src) {
#if defined(__gfx1250__)
  asm volatile("global_load_async_to_lds_b128 %0, %1, off th:TH_LOAD_NT"
               :: "v"((uint32_t)(uintptr_t)lds_dst),
                  "v"((uint64_t)(uintptr_t)g_src)
               : "memory");
#else
  *(v4f*)lds_dst = *(const v4f*)g_src;   // fallback for non-gfx1250 compiles
#endif
}

__device__ __forceinline__ void wait_async0() {
#if defined(__gfx1250__)
#if __has_builtin(__builtin_amdgcn_s_wait_asynccnt)
  __builtin_amdgcn_s_wait_asynccnt(0);
#else
  asm volatile("s_wait_asynccnt 0" ::: "memory");
#endif
#endif
}

__global__ void __launch_bounds__(kBlock)
ln_bwd_fused(const float* __restrict__ dy, const float* __restrict__ x1,
             const float* __restrict__ x2, const float* __restrict__ rstd,
             const float* __restrict__ mean, const float* __restrict__ gamma,
             const float* __restrict__ dsum, float* __restrict__ out) {
  __shared__ float s_dsum[kH];              // async-staged dsum row (8 KB)
  __shared__ float s_red[2][3 * kWaves];    // ping-pong reduction scratch

  const int t = threadIdx.x;
  const int wave = t >> 5;
  const int lane = t & 31;
  const int c0 = t * kCPT;                  // first of 8 contiguous columns

  // gamma tile in registers (reused for all 16 rows of this block)
  const v4f g0 = *(const v4f*)(gamma + c0);
  const v4f g1 = *(const v4f*)(gamma + c0 + 4);

  v4f acc_g0 = {0.f, 0.f, 0.f, 0.f}, acc_g1 = {0.f, 0.f, 0.f, 0.f};
  v4f acc_b0 = {0.f, 0.f, 0.f, 0.f}, acc_b1 = {0.f, 0.f, 0.f, 0.f};

  const int row0 = blockIdx.x * kRowsPerBlock;
  for (int ri = 0; ri < kRowsPerBlock; ++ri) {
    const int r = row0 + ri;
    const int p = ri & 1;                   // ping-pong parity
    const size_t base = (size_t)r * kH + c0;

    // Kick off async dsum staging first; it completes under the reduction.
    async_copy16(&s_dsum[c0],     dsum + base);
    async_copy16(&s_dsum[c0 + 4], dsum + base + 4);

    const float rs = rstd[r];   // uniform -> scalar load
    const float mu = mean[r];

    const v4f dy0 = nt_load4(dy + base), dy1 = nt_load4(dy + base + 4);
    const v4f a0  = nt_load4(x1 + base), a1  = nt_load4(x1 + base + 4);
    const v4f b0  = nt_load4(x2 + base), b1  = nt_load4(x2 + base + 4);

    const v4f xh0 = (a0 + b0) - mu;
    const v4f xh1 = (a1 + b1) - mu;
    const v4f pdxl0 = dy0 * g0;
    const v4f pdxl1 = dy1 * g1;

    // Per-thread partials for the three row reductions.
    const v4f t1 = pdxl0 * xh0 + pdxl1 * xh1;
    const v4f t2 = pdxl0 + pdxl1;
    const v4f t3 = xh0 + xh1;
    float s1 = t1.x + t1.y + t1.z + t1.w;
    float s2 = t2.x + t2.y + t2.z + t2.w;
    float s3 = t3.x + t3.y + t3.z + t3.w;

    // wave32 butterfly reduction (5 steps)
    #pragma unroll
    for (int off = 16; off > 0; off >>= 1) {
      s1 += __shfl_xor(s1, off, 32);
      s2 += __shfl_xor(s2, off, 32);
      s3 += __shfl_xor(s3, off, 32);
    }
    if (lane == 0) {
      s_red[p][wave] = s1;
      s_red[p][kWaves + wave] = s2;
      s_red[p][2 * kWaves + wave] = s3;
    }
    __syncthreads();   // single barrier per row (ping-pong removes the 2nd)
    float S1 = 0.f, S2 = 0.f, S3 = 0.f;
    #pragma unroll
    for (int w = 0; w < kWaves; ++w) {
      S1 += s_red[p][w];
      S2 += s_red[p][kWaves + w];
      S3 += s_red[p][2 * kWaves + w];
    }

    const float r3 = rs * rs * rs;
    const float pd_var = -0.5f * S1 * r3;
    const float pd_mean = -rs * S2 + pd_var * (-kTwoInvD) * S3;
    const float k2 = pd_var * kTwoInvD;
    const float km = pd_mean * kInvD;

    // dsum staged via ASYNCcnt; each thread reads only LDS it wrote itself,
    // so no extra barrier is required around the staging buffer.
    wait_async0();
    const v4f ds0 = *(const v4f*)(&s_dsum[c0]);
    const v4f ds1 = *(const v4f*)(&s_dsum[c0 + 4]);

    nt_store4(out + base,     pdxl0 * rs + k2 * xh0 + km + ds0);
    nt_store4(out + base + 4, pdxl1 * rs + k2 * xh1 + km + ds1);

    // param-grad partials (column ownership fixed across rows)
    acc_g0 += dy0 * xh0 * rs;
    acc_g1 += dy1 * xh1 * rs;
    acc_b0 += dy0;
    acc_b1 += dy1;
  }

  // Flush partials: gamma at out[Rows*H .. +H), beta at out[Rows*H+H .. +2H)
  float* gout = out + (size_t)kRows * kH;
  float* bout = gout + kH;
  #pragma unroll
  for (int j = 0; j < 4; ++j) {
    __hip_atomic_fetch_add(&gout[c0 + j],     acc_g0[j], __ATOMIC_RELAXED, __HIP_MEMORY_SCOPE_AGENT);
    __hip_atomic_fetch_add(&gout[c0 + 4 + j], acc_g1[j], __ATOMIC_RELAXED, __HIP_MEMORY_SCOPE_AGENT);
    __hip_atomic_fetch_add(&bout[c0 + j],     acc_b0[j], __ATOMIC_RELAXED, __HIP_MEMORY_SCOPE_AGENT);
    __hip_atomic_fetch_add(&bout[c0 + 4 + j], acc_b1[j], __ATOMIC_RELAXED, __HIP_MEMORY_SCOPE_AGENT);
  }
}

// Zero the gamma/beta accumulator tail of d_out (must run every call: we
// accumulate with atomics and the harness does not re-poison between replays).
__global__ void zero_tail(float* __restrict__ out) {
  const int i = blockIdx.x * blockDim.x + threadIdx.x;
  if (i < 2 * kH) out[(size_t)kRows * kH + i] = 0.f;
}

extern "C" void kernel_launch(void* const* d_in, const int* in_sizes, int n_in,
                              void* d_out, int out_size, void* d_ws, size_t ws_size,
                              hipStream_t stream) {
  (void)in_sizes; (void)n_in; (void)out_size; (void)d_ws; (void)ws_size;
  const float* dy    = (const float*)d_in[0];
  const float* x1    = (const float*)d_in[1];
  const float* x2    = (const float*)d_in[2];
  const float* rstd  = (const float*)d_in[3];
  const float* mean  = (const float*)d_in[4];
  const float* gamma = (const float*)d_in[5];
  const float* dsum  = (const float*)d_in[6];
  float* out = (float*)d_out;

  zero_tail<<<(2 * kH + kBlock - 1) / kBlock, kBlock, 0, stream>>>(out);
  ln_bwd_fused<<<kGrid, kBlock, 0, stream>>>(dy, x1, x2, rstd, mean, gamma, dsum, out);
}